// BatchRankingMSE_Loss_62191126446112
// MI455X (gfx1250) — compile-verified
//
#include <hip/hip_runtime.h>
#include <math.h>

// BatchRankingMSE loss, N = 8192 (fixed by reference).
//
//   f(i,j) = relu(MARGIN - (p_j - p_i) * sign(l_j - l_i))   (symmetric, f(i,i)=MARGIN)
//   ranking = (sum_{i,j} f - N*MARGIN) / 2
//   grad_k  = sum_j sign(l_j - l_k) * [arg > 0]
//   out     = msesum/N + ((2/N)*sqrt(msesum) / (sqrt(sum grad_k^2) + EPS)) * ranking

#define N_ELEMS 8192
#define MARGIN 2.0f
#define EPSV 1e-4f
#define BLOCK 256
#define JS 8                        // column slices
#define COLS (N_ELEMS / JS)         // 1024 columns per slice
#define ROWBLOCKS (N_ELEMS / BLOCK) // 32 row blocks

// workspace layout (floats)
#define GRAD_OFF 0                           // JS * N partial row-gradients
#define F_OFF (JS * N_ELEMS)                 // ROWBLOCKS*JS = 256 f partials
#define MSE_OFF (F_OFF + ROWBLOCKS * JS)     // 32 mse partials
#define GSQ_OFF (MSE_OFF + ROWBLOCKS)        // 32 gradsq partials

#if defined(__gfx1250__) && __has_builtin(__builtin_amdgcn_global_load_async_to_lds_b128) && \
    __has_builtin(__builtin_amdgcn_s_wait_asynccnt)
#define HAVE_ASYNC_LDS 1
#else
#define HAVE_ASYNC_LDS 0
#endif

#if HAVE_ASYNC_LDS
typedef int v4i __attribute__((ext_vector_type(4)));
typedef __attribute__((address_space(1))) v4i gv4i; // global (AS1) int4
typedef __attribute__((address_space(3))) v4i lv4i; // LDS (AS3) int4
#endif

__device__ __forceinline__ float block_reduce_sum(float v, float* sm) {
  const int tid = threadIdx.x;
  sm[tid] = v;
  __syncthreads();
#pragma unroll
  for (int s = BLOCK / 2; s > 0; s >>= 1) {
    if (tid < s) sm[tid] += sm[tid + s];
    __syncthreads();
  }
  return sm[0];
}

// Grid: (ROWBLOCKS, JS). Each block: rows [bx*256, bx*256+256) x cols [by*1024, ...).
__global__ void __launch_bounds__(BLOCK) pair_kernel(const float* __restrict__ preds,
                                                     const float* __restrict__ labels,
                                                     float* __restrict__ ws) {
  __shared__ __align__(16) float ps[COLS];
  __shared__ __align__(16) float ls[COLS];
  const int tid = threadIdx.x;
  const int bx = blockIdx.x;
  const int by = blockIdx.y;
  const int k = bx * BLOCK + tid;
  const int j0 = by * COLS;

  // Stage this block's column slice into LDS (CDNA5 async global->LDS if available).
#if HAVE_ASYNC_LDS
  {
    __builtin_amdgcn_global_load_async_to_lds_b128(
        (gv4i*)(preds + j0 + tid * 4), (lv4i*)(ps + tid * 4), 0, 0);
    __builtin_amdgcn_global_load_async_to_lds_b128(
        (gv4i*)(labels + j0 + tid * 4), (lv4i*)(ls + tid * 4), 0, 0);
    __builtin_amdgcn_s_wait_asynccnt(0);
  }
#else
  {
    const float4* p4 = (const float4*)(preds + j0);
    const float4* l4 = (const float4*)(labels + j0);
    ((float4*)ps)[tid] = p4[tid];
    ((float4*)ls)[tid] = l4[tid];
  }
#endif
  __syncthreads();

  const float pk = preds[k];
  const float lk = labels[k];

  float fsum = 0.0f;  // sum_j f(k, j) over this column slice
  float g = 0.0f;     // sum_j sign(l_j - l_k) * [arg > 0] over this slice
#pragma unroll 8
  for (int j = 0; j < COLS; ++j) {
    const float pj = ps[j];
    const float lj = ls[j];
    const float s = (lj > lk) ? 1.0f : ((lj < lk) ? -1.0f : 0.0f);
    const float t = fmaf(pk - pj, s, MARGIN); // MARGIN - (pj - pk)*s
    fsum += fmaxf(t, 0.0f);
    g += (t > 0.0f) ? s : 0.0f; // relu' at 0 is 0, matching JAX
  }

  // partial row gradient for this column slice
  ws[GRAD_OFF + by * N_ELEMS + k] = g;

  __syncthreads(); // everyone done reading ps/ls; reuse as reduction scratch
  const float ftot = block_reduce_sum(fsum, ps);
  if (tid == 0) ws[F_OFF + by * ROWBLOCKS + bx] = ftot;

  if (by == 0) { // uniform across block
    const float d = pk - lk;
    const float m = block_reduce_sum(d * d, ls);
    if (tid == 0) ws[MSE_OFF + bx] = m;
  }
}

// Grid: (ROWBLOCKS). Sum grad slices per row, square, block-reduce.
__global__ void __launch_bounds__(BLOCK) gradsq_kernel(float* __restrict__ ws) {
  __shared__ float sm[BLOCK];
  const int k = blockIdx.x * BLOCK + threadIdx.x;
  float gk = 0.0f;
#pragma unroll
  for (int s = 0; s < JS; ++s) gk += ws[GRAD_OFF + s * N_ELEMS + k];
  const float v = block_reduce_sum(gk * gk, sm);
  if (threadIdx.x == 0) ws[GSQ_OFF + blockIdx.x] = v;
}

// One block: fold partials into the scalar loss.
__global__ void __launch_bounds__(BLOCK) final_kernel(const float* __restrict__ ws,
                                                      float* __restrict__ out) {
  __shared__ float sm[BLOCK];
  const int tid = threadIdx.x;
  const float fsum = block_reduce_sum(ws[F_OFF + tid], sm); // exactly 256 partials
  if (tid == 0) {
    float msesum = 0.0f, g2sq = 0.0f;
    for (int i = 0; i < ROWBLOCKS; ++i) {
      msesum += ws[MSE_OFF + i];
      g2sq += ws[GSQ_OFF + i];
    }
    const float n = (float)N_ELEMS;
    const float mse = msesum / n;
    const float g1 = (2.0f / n) * sqrtf(msesum); // ||2(p-l)/N||
    const float g2 = sqrtf(g2sq);
    const float ranking = 0.5f * (fsum - n * MARGIN);
    out[0] = mse + (g1 / (g2 + EPSV)) * ranking;
  }
}

extern "C" void kernel_launch(void* const* d_in, const int* in_sizes, int n_in,
                              void* d_out, int out_size, void* d_ws, size_t ws_size,
                              hipStream_t stream) {
  (void)in_sizes; (void)n_in; (void)out_size; (void)ws_size;
  const float* preds = (const float*)d_in[0];
  const float* labels = (const float*)d_in[1];
  float* ws = (float*)d_ws;
  float* out = (float*)d_out;

  dim3 grid1(ROWBLOCKS, JS);
  pair_kernel<<<grid1, BLOCK, 0, stream>>>(preds, labels, ws);
  gradsq_kernel<<<ROWBLOCKS, BLOCK, 0, stream>>>(ws);
  final_kernel<<<1, BLOCK, 0, stream>>>(ws, out);
}